// EchoStateNetwork_10883447128563
// MI455X (gfx1250) — compile-verified
//
#include <hip/hip_runtime.h>

typedef float v2f __attribute__((ext_vector_type(2)));
typedef float v8f __attribute__((ext_vector_type(8)));
typedef unsigned u32x4 __attribute__((ext_vector_type(4)));
typedef int i32x4 __attribute__((ext_vector_type(4)));
typedef int i32x8 __attribute__((ext_vector_type(8)));

#define B_  32
#define S_  1024
#define I_  128
#define R_  1024
#define O_  128
#define NWG 16

// TDM-padded LDS layout for the W slice: pad_interval = 256 DWORDs (code 7),
// pad_amount = 1 DWORD (code 0) -> each 256-DWORD block occupies 257 DWORDs,
// row stride = 1028 DWORDs (bank step 4 -> conflict-free across 16 lanes).
#define W_BLK_DW   257
#define W_ROW_DW   1028
#define WIN_STRIDE 130     // fused-path W_in slice stride (manual staging)

#define WMMA_F32(a, b, c) \
    __builtin_amdgcn_wmma_f32_16x16x4_f32(false, (a), false, (b), (short)0, (c), false, false)

// ---------------- deterministic noise (PCG hash + Box-Muller) ----------------
__device__ __forceinline__ unsigned pcg_hash(unsigned v) {
    unsigned state = v * 747796405u + 2891336453u;
    unsigned word  = ((state >> ((state >> 28u) + 4u)) ^ state) * 277803737u;
    return (word >> 22u) ^ word;
}
__device__ __forceinline__ float gnoise(unsigned id) {
    unsigned h1 = pcg_hash(id * 2u + 1u);
    unsigned h2 = pcg_hash(id * 2u + 2u);
    float u1 = (float)(h1 >> 8) * (1.0f / 16777216.0f) + 1.0e-7f;
    float u2 = (float)(h2 >> 8) * (1.0f / 16777216.0f);
    return 0.001f * sqrtf(-2.0f * __logf(u1)) * __cosf(6.28318530718f * u2);
}

// ---------------- TDM staging of the 64x1024 fp32 W slice into LDS ----------
__device__ __forceinline__ void tdm_stage_w(const float* W, int wg, float* Wl) {
    if (threadIdx.x < 32) {   // one wave issues the DMA (EXEC ignored by TDM)
        unsigned long long ga = (unsigned long long)(const void*)(W + (size_t)wg * 64 * R_);
        unsigned lds_addr = (unsigned)(unsigned long long)(void*)Wl;  // LDS byte offset (addr[31:0])

        u32x4 g0;
        g0.x = 1u;                                            // count=1 (valid), user mode
        g0.y = lds_addr;                                      // lds_addr
        g0.z = (unsigned)(ga & 0xFFFFFFFFull);                // global_addr[31:0]
        g0.w = (unsigned)((ga >> 32) & 0x01FFFFFFull)         // global_addr[56:32]
             | (2u << 30);                                    // type=2 ("image")

        i32x8 g1;
        g1[0] = (int)((2u << 16)     // data_size = 4 bytes
                    | (1u << 20)     // pad_enable
                    | (7u << 22)     // pad_interval: 256 DWORDs
                    | (0u << 25));   // pad_amount: 1 DWORD
        g1[1] = (int)(((unsigned)R_ & 0xFFFFu) << 16);        // barrier_addr=0 | tensor_dim0 lo16
        g1[2] = (int)((((unsigned)R_ >> 16) & 0xFFFFu)        // tensor_dim0 hi16
                    | (((unsigned)R_ & 0xFFFFu) << 16));      // tensor_dim1 lo16
        g1[3] = (int)((((unsigned)R_ >> 16) & 0xFFFFu)        // tensor_dim1 hi16
                    | ((unsigned)R_ << 16));                  // tile_dim0 = 1024
        g1[4] = 64;                                           // tile_dim1 = 64, tile_dim2 = 0
        g1[5] = R_;                                           // tensor_dim0_stride = 1024 (lo32)
        g1[6] = 0;
        g1[7] = 0;

        i32x4 gz = {0, 0, 0, 0};
#if defined(__clang_major__) && (__clang_major__ >= 23)
        i32x8 gz8 = {0, 0, 0, 0, 0, 0, 0, 0};
        __builtin_amdgcn_tensor_load_to_lds(g0, g1, gz, gz, gz8, 0);
#else
        __builtin_amdgcn_tensor_load_to_lds(g0, g1, gz, gz, 0);
#endif
        __builtin_amdgcn_s_wait_tensorcnt((short)0);
    }
}

// ---------------- barrier init ----------------
__global__ void esn_bar_init(int* bar) { bar[0] = 0; bar[1] = 0; }

// ---------------- parallel pre-pass: xpn[s] = x[:,s,:] @ W_in^T + noise -----
__global__ __launch_bounds__(256)
void esn_xproj(const float* __restrict__ x,    // [B,S,I]
               const float* __restrict__ Win,  // [R,I]
               float* __restrict__ xpn)        // [S,B,R]
{
    const int s    = blockIdx.x >> 4;
    const int wg   = blockIdx.x & 15;
    const int tid  = threadIdx.x;
    const int lane = tid & 31;
    const int wave = tid >> 5;
    const int mt   = wave & 1;
    const int nt   = wave >> 1;
    const int half = lane >> 4;
    const int l15  = lane & 15;

    const int m_row = mt * 16 + l15;
    const int r_col = wg * 64 + nt * 16 + l15;

    const float* ap = x   + (size_t)m_row * (S_ * I_) + (size_t)s * I_ + half * 2;
    const float* wb = Win + (size_t)r_col * I_ + half * 2;

    v8f c0 = {}, c1 = {};
    #pragma unroll
    for (int k = 0; k < I_; k += 8) {
        v2f a0 = *(const v2f*)(ap + k);
        v2f b0 = *(const v2f*)(wb + k);
        c0 = WMMA_F32(a0, b0, c0);
        v2f a1 = *(const v2f*)(ap + k + 4);
        v2f b1 = *(const v2f*)(wb + k + 4);
        c1 = WMMA_F32(a1, b1, c1);
    }

    float* op = xpn + (size_t)s * (B_ * R_);
    #pragma unroll
    for (int v = 0; v < 8; ++v) {
        const int b = mt * 16 + v + 8 * half;
        const unsigned id = ((unsigned)s * B_ + (unsigned)b) * R_ + (unsigned)r_col;
        op[(size_t)b * R_ + r_col] = c0[v] + c1[v] + gnoise(id);
    }
}

// ---------------- persistent recurrence kernel (16 WGs, W slice in LDS) -----
template <bool FUSED>
__global__ __launch_bounds__(256)
void esn_recur(const float* __restrict__ x,    // [B,S,I]   (FUSED only)
               const float* __restrict__ Win,  // [R,I]     (FUSED only)
               const float* __restrict__ W,    // [R,R]
               const float* __restrict__ xpn,  // [S,B,R]   (!FUSED only)
               float* __restrict__ h_all,      // [S,B,R]
               int* __restrict__ bar)
{
    extern __shared__ float smem[];
    float* Wl   = smem;                  // [64][W_ROW_DW] via TDM padding
    float* WinL = smem + 64 * W_ROW_DW;  // [64][WIN_STRIDE] (FUSED only)

    const int tid  = threadIdx.x;
    const int wg   = blockIdx.x;   // 0..15
    const int lane = tid & 31;
    const int wave = tid >> 5;     // 0..7
    const int mt   = wave & 1;
    const int nt   = wave >> 1;
    const int half = lane >> 4;
    const int l15  = lane & 15;

    tdm_stage_w(W, wg, Wl);        // async-tensor DMA, wave 0 waits tensorcnt

    if (FUSED) {
        const int row = tid >> 2;
        const int kq2 = (tid & 3) * 32;
        const float4* src2 = (const float4*)(Win + (size_t)(wg * 64 + row) * I_ + kq2);
        float* dst2 = WinL + row * WIN_STRIDE + kq2;
        #pragma unroll
        for (int i = 0; i < 8; ++i) {
            float4 v = src2[i];
            dst2[i*4+0] = v.x; dst2[i*4+1] = v.y; dst2[i*4+2] = v.z; dst2[i*4+3] = v.w;
        }
    }
    __syncthreads();

    const int m_row = mt * 16 + l15;
    const int n_loc = nt * 16 + l15;
    const int r_col = wg * 64 + n_loc;
    const float* wrow = Wl   + n_loc * W_ROW_DW  + half * 2;
    const float* winb = WinL + n_loc * WIN_STRIDE + half * 2;

    for (int s = 0; s < S_; ++s) {
        v8f c0 = {}, c1 = {};

        if (FUSED) {  // input projection fused into the step (K = 128)
            const float* ap = x + (size_t)m_row * (S_ * I_) + (size_t)s * I_ + half * 2;
            #pragma unroll
            for (int k = 0; k < I_; k += 8) {
                v2f a0 = *(const v2f*)(ap + k);
                v2f b0; b0.x = winb[k];     b0.y = winb[k + 1];
                c0 = WMMA_F32(a0, b0, c0);
                v2f a1 = *(const v2f*)(ap + k + 4);
                v2f b1; b1.x = winb[k + 4]; b1.y = winb[k + 5];
                c1 = WMMA_F32(a1, b1, c1);
            }
        }

        if (s > 0) {  // recurrent term, K = 1024 in 4 TDM-padded 256-DW blocks
            const float* hp = h_all + (size_t)(s - 1) * (B_ * R_) + (size_t)m_row * R_ + half * 2;
            #pragma unroll
            for (int blk = 0; blk < 4; ++blk) {
                const float* hb = hp   + blk * 256;
                const float* wb = wrow + blk * W_BLK_DW;
                #pragma unroll 4
                for (int k = 0; k < 256; k += 8) {
                    v2f a0 = *(const v2f*)(hb + k);
                    v2f b0; b0.x = wb[k];     b0.y = wb[k + 1];
                    c0 = WMMA_F32(a0, b0, c0);
                    v2f a1 = *(const v2f*)(hb + k + 4);
                    v2f b1; b1.x = wb[k + 4]; b1.y = wb[k + 5];
                    c1 = WMMA_F32(a1, b1, c1);
                }
            }
        }

        // h_new = tanh(acc + xpn)  (xpn already contains x-proj + noise)
        float* hout = h_all + (size_t)s * (B_ * R_);
        #pragma unroll
        for (int v = 0; v < 8; ++v) {
            const int b = mt * 16 + v + 8 * half;
            float pre = c0[v] + c1[v];
            if (FUSED) {
                const unsigned id = ((unsigned)s * B_ + (unsigned)b) * R_ + (unsigned)r_col;
                pre += gnoise(id);
            } else {
                pre += xpn[(size_t)s * (B_ * R_) + (size_t)b * R_ + r_col];
            }
            hout[(size_t)b * R_ + r_col] = tanhf(pre);
        }

        // device-wide step barrier (16 co-resident WGs)
        if (s + 1 < S_) {
            __threadfence();
            __syncthreads();
            if (tid == 0) {
                int arrive = __hip_atomic_fetch_add(&bar[0], 1, __ATOMIC_ACQ_REL,
                                                    __HIP_MEMORY_SCOPE_AGENT);
                if (arrive == NWG - 1) {
                    __hip_atomic_store(&bar[0], 0, __ATOMIC_RELAXED, __HIP_MEMORY_SCOPE_AGENT);
                    __hip_atomic_fetch_add(&bar[1], 1, __ATOMIC_RELEASE, __HIP_MEMORY_SCOPE_AGENT);
                } else {
                    while (__hip_atomic_load(&bar[1], __ATOMIC_ACQUIRE,
                                             __HIP_MEMORY_SCOPE_AGENT) <= s) {
                        __builtin_amdgcn_s_sleep(1);
                    }
                }
            }
            __builtin_amdgcn_s_cluster_barrier();  // NOP unless cluster-dispatched
            __syncthreads();
        }
    }
}

// ---------------- parallel readout: y[s] = h_all[s] @ W_out^T + b -----------
__global__ __launch_bounds__(256)
void esn_readout(const float* __restrict__ h_all,  // [S,B,R]
                 const float* __restrict__ Wout,   // [O,R]
                 const float* __restrict__ bout,   // [O]
                 float* __restrict__ out)          // [B,S,O]
{
    const int s    = blockIdx.x;
    const int tid  = threadIdx.x;
    const int lane = tid & 31;
    const int nt   = tid >> 5;
    const int half = lane >> 4;
    const int l15  = lane & 15;
    const int o    = nt * 16 + l15;

    const float* hp0 = h_all + (size_t)s * (B_ * R_) + (size_t)l15 * R_ + half * 2;
    const float* hp1 = hp0 + 16 * R_;
    const float* wb  = Wout + (size_t)o * R_ + half * 2;

    v8f c0 = {}, c1 = {};
    #pragma unroll 4
    for (int k = 0; k < R_; k += 4) {
        v2f b  = *(const v2f*)(wb + k);
        v2f a0 = *(const v2f*)(hp0 + k);
        v2f a1 = *(const v2f*)(hp1 + k);
        c0 = WMMA_F32(a0, b, c0);
        c1 = WMMA_F32(a1, b, c1);
    }

    const float bias = bout[o];
    #pragma unroll
    for (int v = 0; v < 8; ++v) {
        const int m = v + 8 * half;
        out[(size_t)m        * (S_ * O_) + (size_t)s * O_ + o] = c0[v] + bias;
        out[(size_t)(16 + m) * (S_ * O_) + (size_t)s * O_ + o] = c1[v] + bias;
    }
}

// ---------------- launcher ----------------
extern "C" void kernel_launch(void* const* d_in, const int* in_sizes, int n_in,
                              void* d_out, int out_size, void* d_ws, size_t ws_size,
                              hipStream_t stream) {
    (void)in_sizes; (void)n_in; (void)out_size;
    const float* x   = (const float*)d_in[0];   // [B,S,I]
    const float* Win = (const float*)d_in[1];   // [R,I]
    const float* W   = (const float*)d_in[2];   // [R,R]
    const float* Ww  = (const float*)d_in[3];   // [O,R]
    const float* Wb  = (const float*)d_in[4];   // [O]
    float* out = (float*)d_out;                 // [B,S,O]

    const size_t hbytes = (size_t)S_ * B_ * R_ * sizeof(float);   // 128 MB
    float* h_all = (float*)d_ws;
    const bool precompute = (ws_size >= 2 * hbytes + 256);

    if (precompute) {
        float* xpn = (float*)((char*)d_ws + hbytes);
        int*   bar = (int*)((char*)d_ws + 2 * hbytes);
        esn_bar_init<<<1, 1, 0, stream>>>(bar);
        esn_xproj<<<S_ * 16, 256, 0, stream>>>(x, Win, xpn);
        const size_t lds_bytes = (size_t)64 * W_ROW_DW * sizeof(float);            // ~257 KB
        esn_recur<false><<<NWG, 256, lds_bytes, stream>>>(x, Win, W, xpn, h_all, bar);
    } else {
        int* bar = (int*)((char*)d_ws + hbytes);
        esn_bar_init<<<1, 1, 0, stream>>>(bar);
        const size_t lds_bytes = (size_t)(64 * W_ROW_DW + 64 * WIN_STRIDE) * sizeof(float); // ~290 KB
        esn_recur<true><<<NWG, 256, lds_bytes, stream>>>(x, Win, W, nullptr, h_all, bar);
    }

    esn_readout<<<S_, 256, 0, stream>>>(h_all, Ww, Wb, out);
}